// MambaBlocksWrapper_32968168964658
// MI455X (gfx1250) — compile-verified
//
#include <hip/hip_runtime.h>
#include <hip/hip_bf16.h>
#include <math.h>

// ---------------- problem constants ----------------
#define BATCH      8
#define SEQLEN     4096
#define DMODEL     512
#define DINNER     1024
#define DSTATE     16
#define DTRANK     32
#define DCONV      4
#define NLAYERS    4
#define NTOK       (BATCH * SEQLEN)          // 32768
#define EPSV       1e-5f

typedef __attribute__((ext_vector_type(2))) float v2f;
typedef __attribute__((ext_vector_type(8))) float v8f;

// ---- CDNA5 async global->LDS (ASYNCcnt path), via inline asm ----
// the vdst VGPR holds the LDS byte address; low 32 bits of a generic pointer to
// __shared__ give the LDS byte offset (flat LDS aperture keeps offset in [31:0]).
__device__ __forceinline__ void async_copy_b128(void* lds_dst, const float* gsrc)
{
    unsigned lds = (unsigned)(uintptr_t)lds_dst;
    asm volatile("global_load_async_to_lds_b128 %0, %1, off"
                 :: "v"(lds), "v"(gsrc) : "memory");
}
__device__ __forceinline__ void wait_async0()
{
    asm volatile("s_wait_asynccnt 0x0" ::: "memory");
}

// ---------------- elementwise: h = x + pos ----------------
__global__ __launch_bounds__(256)
void add_kernel(const float* __restrict__ x, const float* __restrict__ p,
                float* __restrict__ out, int n)
{
    int i = blockIdx.x * 256 + threadIdx.x;
    if (i < n) out[i] = x[i] + p[i];
}

// ---------------- rmsnorm: one wave32 per token (512 feats) ----------------
__global__ __launch_bounds__(256)
void rmsnorm_kernel(const float* __restrict__ x, const float* __restrict__ w,
                    float* __restrict__ out)
{
    const int wid  = threadIdx.x >> 5;
    const int lane = threadIdx.x & 31;
    const int tok  = blockIdx.x * 8 + wid;
    const float* xr = x + (size_t)tok * DMODEL;

    float v[16];
    float ss = 0.f;
#pragma unroll
    for (int j = 0; j < 16; ++j) {
        v[j] = xr[lane + 32 * j];
        ss = fmaf(v[j], v[j], ss);
    }
#pragma unroll
    for (int off = 16; off > 0; off >>= 1)
        ss += __shfl_xor(ss, off, 32);
    const float s = rsqrtf(ss * (1.f / DMODEL) + EPSV);
    float* orow = out + (size_t)tok * DMODEL;
#pragma unroll
    for (int j = 0; j < 16; ++j) {
        int c = lane + 32 * j;
        orow[c] = v[j] * s * w[c];
    }
}

// ---------------- WMMA fp32 GEMM:  C[M,N] = A[M,K] * W[N,K]^T (+Res) -------
// 256 thr = 8 wave32; block tile 128x64; wave tile 32x32; K-chunk 32.
// Double-buffered LDS filled by async global->LDS b128 DMA.
#define BM 128
#define BN 64
#define BK 32
#define LSTR 36   // padded LDS row stride in floats (multiple of 4 for b128)

__global__ __launch_bounds__(256)
void gemm_wmma_f32(const float* __restrict__ A, const float* __restrict__ W,
                   float* __restrict__ C, const float* __restrict__ Res,
                   int K, int lda, int ldc)
{
    __shared__ float As[2][BM * LSTR];   // 2 x 18 KB
    __shared__ float Ws[2][BN * LSTR];   // 2 x  9 KB

    const int tid  = threadIdx.x;
    const int wid  = tid >> 5;
    const int lane = tid & 31;
    const int lo   = lane & 15;
    const int hi   = lane >> 4;          // 0 or 1

    const int m0 = blockIdx.y * BM;
    const int n0 = blockIdx.x * BN;
    const int wm = (wid & 3) * 32;       // wave row offset in block tile
    const int wn = (wid >> 2) * 32;      // wave col offset in block tile

    v8f acc[2][2] = {};

    // stage one 128x32 A chunk + 64x32 W chunk into buffer `buf` via async DMA
    auto stage = [&](int buf, int k0) {
        // A: 1024 float4 -> 4 per thread
#pragma unroll
        for (int j = 0; j < 4; ++j) {
            int idx = tid + j * 256;
            int r = idx >> 3, c4 = idx & 7;            // 8 float4 per row
            async_copy_b128(&As[buf][r * LSTR + 4 * c4],
                            &A[(size_t)(m0 + r) * lda + k0 + 4 * c4]);
        }
        // W: 512 float4 -> 2 per thread
#pragma unroll
        for (int j = 0; j < 2; ++j) {
            int idx = tid + j * 256;
            int r = idx >> 3, c4 = idx & 7;
            async_copy_b128(&Ws[buf][r * LSTR + 4 * c4],
                            &W[(size_t)(n0 + r) * K + k0 + 4 * c4]);
        }
    };

    stage(0, 0);
    wait_async0();
    __syncthreads();

    int buf = 0;
    for (int k0 = 0; k0 < K; k0 += BK) {
        if (k0 + BK < K) stage(buf ^ 1, k0 + BK);   // overlap DMA with WMMA

        const float* as = &As[buf][0];
        const float* wsb = &Ws[buf][0];
#pragma unroll
        for (int kk = 0; kk < BK / 4; ++kk) {
            const int kc = kk * 4 + 2 * hi;          // lanes 0-15: K={0,1}; 16-31: K={2,3}
            v2f af[2], bf[2];
#pragma unroll
            for (int mi = 0; mi < 2; ++mi) {
                af[mi][0] = as[(wm + 16 * mi + lo) * LSTR + kc];
                af[mi][1] = as[(wm + 16 * mi + lo) * LSTR + kc + 1];
            }
#pragma unroll
            for (int ni = 0; ni < 2; ++ni) {
                bf[ni][0] = wsb[(wn + 16 * ni + lo) * LSTR + kc];
                bf[ni][1] = wsb[(wn + 16 * ni + lo) * LSTR + kc + 1];
            }
#pragma unroll
            for (int mi = 0; mi < 2; ++mi)
#pragma unroll
                for (int ni = 0; ni < 2; ++ni)
                    acc[mi][ni] = __builtin_amdgcn_wmma_f32_16x16x4_f32(
                        false, af[mi], false, bf[ni], (short)0, acc[mi][ni],
                        false, false);
        }

        wait_async0();        // next buffer's DMA done
        __syncthreads();      // everyone finished reading current buffer
        buf ^= 1;
    }

    // D layout: VGPR r -> row (r + 8*hi), col = lo  (per 16x16 tile)
#pragma unroll
    for (int mi = 0; mi < 2; ++mi)
#pragma unroll
        for (int ni = 0; ni < 2; ++ni)
#pragma unroll
            for (int r = 0; r < 8; ++r) {
                const int row = m0 + wm + 16 * mi + 8 * hi + r;
                const int col = n0 + wn + 16 * ni + lo;
                float v = acc[mi][ni][r];
                if (Res) v += Res[(size_t)row * ldc + col];
                C[(size_t)row * ldc + col] = v;
            }
}

// ---------------- depthwise causal conv (k=4) + SiLU ----------------
// reads xi half of xz (stride 2048), writes xc (stride 1024)
__global__ __launch_bounds__(256)
void conv_silu_kernel(const float* __restrict__ xz, const float* __restrict__ cw,
                      const float* __restrict__ cb, float* __restrict__ xc)
{
    const size_t idx = (size_t)blockIdx.x * 256 + threadIdx.x; // over NTOK*DINNER
    const int    e   = (int)(idx & (DINNER - 1));
    const size_t tok = idx >> 10;
    const int    t   = (int)(tok & (SEQLEN - 1));

    float acc = cb[e];
#pragma unroll
    for (int j = 0; j < DCONV; ++j) {
        const int tt = t + j - (DCONV - 1);
        if (tt >= 0)
            acc = fmaf(cw[e * DCONV + j], xz[(tok + j - (DCONV - 1)) * 2048 + e], acc);
    }
    xc[idx] = acc / (1.f + expf(-acc));   // silu
}

// ---------------- fused dt_proj + softplus + selective scan + gate ---------
// one thread per (b, e); 16 states in registers; dbc rows staged in LDS via
// double-buffered async DMA so the next chunk streams in during the scan.
__global__ __launch_bounds__(256)
void scan_kernel(const float* __restrict__ dbc,    // (NTOK, 64)
                 const float* __restrict__ xc,     // (NTOK, DINNER)
                 float* __restrict__ xz,           // (NTOK, 2048): z at +1024, y out at +0
                 const float* __restrict__ dtw,    // (DINNER, DTRANK)
                 const float* __restrict__ dtb,    // (DINNER)
                 const float* __restrict__ A_log,  // (DINNER, DSTATE)
                 const float* __restrict__ Dsk)    // (DINNER)
{
    __shared__ float sbuf[2][64][64];              // 2 x 16 KB: 64 t x (32 dlt|16 B|16 C)

    const int tid = threadIdx.x;
    const int b   = blockIdx.x >> 2;
    const int e   = (blockIdx.x & 3) * 256 + tid;

    float An[DSTATE];
#pragma unroll
    for (int n = 0; n < DSTATE; ++n) An[n] = -expf(A_log[e * DSTATE + n]);
    float wr[DTRANK];
#pragma unroll
    for (int r = 0; r < DTRANK; ++r) wr[r] = dtw[e * DTRANK + r];
    const float bias = dtb[e];
    const float dsk  = Dsk[e];

    float h[DSTATE];
#pragma unroll
    for (int n = 0; n < DSTATE; ++n) h[n] = 0.f;

    const size_t tokbase = (size_t)b * SEQLEN;

    // chunk is 4096 contiguous floats = 1024 float4 -> 4 async b128 per thread
    auto stage = [&](int bufi, int t0) {
#pragma unroll
        for (int j = 0; j < 4; ++j) {
            int idx = tid + j * 256;
            async_copy_b128(&sbuf[bufi][0][0] + 4 * idx,
                            &dbc[(tokbase + t0) * 64 + 4 * idx]);
        }
    };

    stage(0, 0);
    wait_async0();
    __syncthreads();

    int bufi = 0;
    for (int t0 = 0; t0 < SEQLEN; t0 += 64) {
        if (t0 + 64 < SEQLEN) stage(bufi ^ 1, t0 + 64);

        for (int tt = 0; tt < 64; ++tt) {
            const size_t tok = tokbase + t0 + tt;
            // delta = softplus(dlt . dt_w[e] + dt_b[e])
            float draw = bias;
#pragma unroll
            for (int r = 0; r < DTRANK; ++r) draw = fmaf(sbuf[bufi][tt][r], wr[r], draw);
            const float delta = (draw > 20.f) ? draw : log1pf(expf(draw));

            const float xin = xc[tok * DINNER + e];
            float y = 0.f;
#pragma unroll
            for (int n = 0; n < DSTATE; ++n) {
                const float dA = expf(delta * An[n]);
                h[n] = fmaf(dA, h[n], (delta * sbuf[bufi][tt][DTRANK + n]) * xin);
                y = fmaf(h[n], sbuf[bufi][tt][DTRANK + DSTATE + n], y);
            }
            const float z = xz[tok * 2048 + DINNER + e];
            const float g = z / (1.f + expf(-z));          // silu(z)
            xz[tok * 2048 + e] = (y + xin * dsk) * g;      // overwrite dead xi slot
        }

        wait_async0();
        __syncthreads();
        bufi ^= 1;
    }
}

// ---------------- host-side orchestration ----------------
extern "C" void kernel_launch(void* const* d_in, const int* in_sizes, int n_in,
                              void* d_out, int out_size, void* d_ws, size_t ws_size,
                              hipStream_t stream)
{
    const float* x     = (const float*)d_in[0];
    const float* pos   = (const float*)d_in[1];
    const float* nrm_w = (const float*)d_in[2];
    const float* inp_w = (const float*)d_in[3];
    const float* cw    = (const float*)d_in[4];
    const float* cb    = (const float*)d_in[5];
    const float* xp_w  = (const float*)d_in[6];
    const float* dt_w  = (const float*)d_in[7];
    const float* dt_b  = (const float*)d_in[8];
    const float* A_log = (const float*)d_in[9];
    const float* Dsk   = (const float*)d_in[10];
    const float* out_w = (const float*)d_in[11];

    float* h = (float*)d_out;                       // (NTOK, 512), updated in place

    // workspace layout (floats)
    float* ws  = (float*)d_ws;
    float* xn  = ws;                                        // (NTOK, 512)
    float* xz  = xn  + (size_t)NTOK * DMODEL;               // (NTOK, 2048)
    float* xc  = xz  + (size_t)NTOK * 2 * DINNER;           // (NTOK, 1024)
    float* dbc = xc  + (size_t)NTOK * DINNER;               // (NTOK, 64)

    // h = x + pos
    {
        const int n = NTOK * DMODEL;
        add_kernel<<<(n + 255) / 256, 256, 0, stream>>>(x, pos, h, n);
    }

    for (int i = 0; i < NLAYERS; ++i) {
        const float* inp_wi = inp_w + (size_t)i * 2 * DINNER * DMODEL;
        const float* cwi    = cw    + (size_t)i * DINNER * DCONV;
        const float* cbi    = cb    + (size_t)i * DINNER;
        const float* xp_wi  = xp_w  + (size_t)i * (DTRANK + 2 * DSTATE) * DINNER;
        const float* dt_wi  = dt_w  + (size_t)i * DINNER * DTRANK;
        const float* dt_bi  = dt_b  + (size_t)i * DINNER;
        const float* A_li   = A_log + (size_t)i * DINNER * DSTATE;
        const float* Dski   = Dsk   + (size_t)i * DINNER;
        const float* out_wi = out_w + (size_t)i * DMODEL * DINNER;
        const float* nrm_wi = nrm_w + (size_t)i * DMODEL;

        // xn = rmsnorm(h)
        rmsnorm_kernel<<<NTOK / 8, 256, 0, stream>>>(h, nrm_wi, xn);

        // xz = xn @ in_proj_w^T   (M=32768, N=2048, K=512)
        {
            dim3 g((2 * DINNER) / BN, NTOK / BM);
            gemm_wmma_f32<<<g, 256, 0, stream>>>(xn, inp_wi, xz, nullptr,
                                                 DMODEL, DMODEL, 2 * DINNER);
        }

        // xc = silu(causal depthwise conv(xi))
        {
            const size_t n = (size_t)NTOK * DINNER;
            conv_silu_kernel<<<(unsigned)(n / 256), 256, 0, stream>>>(xz, cwi, cbi, xc);
        }

        // dbc = xc @ x_proj_w^T   (M=32768, N=64, K=1024)
        {
            dim3 g((DTRANK + 2 * DSTATE) / BN, NTOK / BM);
            gemm_wmma_f32<<<g, 256, 0, stream>>>(xc, xp_wi, dbc, nullptr,
                                                 DINNER, DINNER, DTRANK + 2 * DSTATE);
        }

        // fused dt_proj + softplus + selective scan + D-skip + silu(z) gate
        scan_kernel<<<BATCH * DINNER / 256, 256, 0, stream>>>(
            dbc, xc, xz, dt_wi, dt_bi, A_li, Dski);

        // h += y @ out_proj_w^T   (A = y stored in xz with lda=2048; residual epilogue)
        {
            dim3 g(DMODEL / BN, NTOK / BM);
            gemm_wmma_f32<<<g, 256, 0, stream>>>(xz, out_wi, h, h,
                                                 DINNER, 2 * DINNER, DMODEL);
        }
    }
}